// gromov_dtw_13056700580131
// MI455X (gfx1250) — compile-verified
//
#include <hip/hip_runtime.h>

typedef __attribute__((ext_vector_type(2))) float v2f;
typedef __attribute__((ext_vector_type(8))) float v8f;

#define NDIM 1024
#define GAMMA 0.5f
#define INVG 2.0f
#define BIGV 1e10f

// ---------------- transpose (1024x64 -> 64x1024) ----------------
__global__ void k_transpose(const float* __restrict__ X, float* __restrict__ XT,
                            int rows, int cols) {
    int idx = blockIdx.x * blockDim.x + threadIdx.x;
    if (idx >= rows * cols) return;
    int r = idx / cols, c = idx % cols;
    XT[c * rows + r] = X[r * cols + c];
}

// ---------------- f32 WMMA GEMM: C = A[MxK] @ B[KxN] ----------------
// Wave computes a 32(M) x 64(N) tile: 2 A-frags x 4 B-frags -> 8 WMMAs / 4-k-step.
// gridDim.y = #K-splits; split s handles K-range [s*K/ns, (s+1)*K/ns) and writes
// its partial result at C + s*M*N (ns==1 writes the final result directly).
// A-frag (16x4, v2f): lane l -> m = l&15, k-half = (l>>4)*2
// B-frag (4x16, v2f): lane l -> n = l&15, k-half = (l>>4)*2
// D-frag (16x16, v8f): lane l, reg r -> m = r + 8*(l>>4), n = l&15
__global__ __launch_bounds__(256) void k_gemm_wmma(const float* __restrict__ A,
                                                   const float* __restrict__ B,
                                                   float* __restrict__ C,
                                                   int M, int N, int K) {
    const int lane = threadIdx.x & 31;
    const int wave = threadIdx.x >> 5;
    const int tilesN = N >> 6;                  // N/64
    const int tilesTotal = (M >> 5) * tilesN;   // (M/32)*(N/64)
    int tileId = blockIdx.x * 8 + wave;
    if (tileId >= tilesTotal) return;
    const int tm = (tileId / tilesN) << 5;
    const int tn = (tileId % tilesN) << 6;

    const int kChunk = K / gridDim.y;
    const int k0 = blockIdx.y * kChunk;
    const int steps = kChunk >> 2;              // k-steps of 4
    float* Cout = C + (size_t)blockIdx.y * M * N;

    const int nm    = lane & 15;                // m for A-frag, n for B-frag
    const int khalf = (lane >> 4) << 1;         // 0 or 2

    v8f c00 = {}, c01 = {}, c02 = {}, c03 = {};
    v8f c10 = {}, c11 = {}, c12 = {}, c13 = {};

    // strength-reduced pointers: +4 floats (A) / +4 rows (B) per k-step
    const float* pA0 = A + (size_t)(tm + nm) * K + khalf + k0;
    const float* pA1 = pA0 + (size_t)16 * K;
    const float* pB0 = B + (size_t)(k0 + khalf) * N + tn + nm;   // row k+khalf
    const float* pB1 = pB0 + N;                                  // row k+khalf+1
    const size_t bStep = (size_t)4 * N;

#pragma unroll 2
    for (int s = 0; s < steps; ++s) {
        v2f a0, a1;
        a0.x = pA0[0];
        a0.y = pA0[1];
        a1.x = pA1[0];
        a1.y = pA1[1];
        v2f bb0, bb1, bb2, bb3;
        bb0.x = pB0[0];  bb0.y = pB1[0];
        bb1.x = pB0[16]; bb1.y = pB1[16];
        bb2.x = pB0[32]; bb2.y = pB1[32];
        bb3.x = pB0[48]; bb3.y = pB1[48];
        c00 = __builtin_amdgcn_wmma_f32_16x16x4_f32(false, a0, false, bb0, (short)0, c00, false, false);
        c01 = __builtin_amdgcn_wmma_f32_16x16x4_f32(false, a0, false, bb1, (short)0, c01, false, false);
        c02 = __builtin_amdgcn_wmma_f32_16x16x4_f32(false, a0, false, bb2, (short)0, c02, false, false);
        c03 = __builtin_amdgcn_wmma_f32_16x16x4_f32(false, a0, false, bb3, (short)0, c03, false, false);
        c10 = __builtin_amdgcn_wmma_f32_16x16x4_f32(false, a1, false, bb0, (short)0, c10, false, false);
        c11 = __builtin_amdgcn_wmma_f32_16x16x4_f32(false, a1, false, bb1, (short)0, c11, false, false);
        c12 = __builtin_amdgcn_wmma_f32_16x16x4_f32(false, a1, false, bb2, (short)0, c12, false, false);
        c13 = __builtin_amdgcn_wmma_f32_16x16x4_f32(false, a1, false, bb3, (short)0, c13, false, false);
        pA0 += 4;
        pA1 += 4;
        pB0 += bStep;
        pB1 += bStep;
    }

    const int mhi = (lane >> 4) << 3;
    float* Cc = Cout + tn + nm;
#pragma unroll
    for (int r = 0; r < 8; ++r) {
        size_t m0 = (size_t)(tm + r + mhi) * N;
        size_t m1 = (size_t)(tm + 16 + r + mhi) * N;
        Cc[m0 + 0]  = c00[r];
        Cc[m0 + 16] = c01[r];
        Cc[m0 + 32] = c02[r];
        Cc[m0 + 48] = c03[r];
        Cc[m1 + 0]  = c10[r];
        Cc[m1 + 16] = c11[r];
        Cc[m1 + 32] = c12[r];
        Cc[m1 + 48] = c13[r];
    }
}

// ---------------- sum of two K-split partials ----------------
__global__ void k_add2(const float* __restrict__ P, float* __restrict__ Cdst, size_t nn) {
    size_t idx = (size_t)blockIdx.x * blockDim.x + threadIdx.x;
    if (idx < nn) Cdst[idx] = P[idx] + P[nn + idx];
}

// ---------------- norms from Gram diagonal ----------------
__global__ void k_diag(const float* __restrict__ G, float* __restrict__ nrm, int n) {
    int i = blockIdx.x * blockDim.x + threadIdx.x;
    if (i < n) nrm[i] = G[(size_t)i * n + i];
}

// ---------------- cost finish: C = n_i + n_j - 2G (in place), F = C*C ----------------
__global__ void k_cost_finish(float* __restrict__ G, float* __restrict__ F,
                              const float* __restrict__ nrm, int n) {
    int j = blockIdx.x * blockDim.x + threadIdx.x;
    int i = blockIdx.y;
    size_t idx = (size_t)i * n + j;
    float c = nrm[i] + nrm[j] - 2.0f * G[idx];
    G[idx] = c;
    F[idx] = c * c;
}

__global__ void k_identity(float* __restrict__ A, int n) {
    int j = blockIdx.x * blockDim.x + threadIdx.x;
    int i = blockIdx.y;
    A[(size_t)i * n + j] = (i == j) ? 1.0f : 0.0f;
}

// ---------------- row sum / col sum / matvec ----------------
__global__ __launch_bounds__(256) void k_rowsum(const float* __restrict__ A,
                                                float* __restrict__ p, int n) {
    __shared__ float s[256];
    int row = blockIdx.x;
    float acc = 0.0f;
    for (int j = threadIdx.x; j < n; j += 256) acc += A[(size_t)row * n + j];
    s[threadIdx.x] = acc; __syncthreads();
    for (int off = 128; off > 0; off >>= 1) {
        if (threadIdx.x < off) s[threadIdx.x] += s[threadIdx.x + off];
        __syncthreads();
    }
    if (threadIdx.x == 0) p[row] = s[0];
}

__global__ void k_colsum_part(const float* __restrict__ A, float* __restrict__ part, int n) {
    int j = blockIdx.x * blockDim.x + threadIdx.x;   // column
    int chunk = blockIdx.y;                          // 8 chunks of n/8 rows
    int r0 = chunk * (n >> 3);
    float acc = 0.0f;
    for (int i = r0; i < r0 + (n >> 3); ++i) acc += A[(size_t)i * n + j];
    part[(size_t)chunk * n + j] = acc;
}

__global__ void k_colsum_final(const float* __restrict__ part, float* __restrict__ q, int n) {
    int j = blockIdx.x * blockDim.x + threadIdx.x;
    float acc = 0.0f;
#pragma unroll
    for (int c = 0; c < 8; ++c) acc += part[(size_t)c * n + j];
    q[j] = acc;
}

__global__ __launch_bounds__(256) void k_matvec(const float* __restrict__ F,
                                                const float* __restrict__ v,
                                                float* __restrict__ o, int n) {
    __shared__ float s[256];
    int row = blockIdx.x;
    float acc = 0.0f;
    for (int j = threadIdx.x; j < n; j += 256) acc += F[(size_t)row * n + j] * v[j];
    s[threadIdx.x] = acc; __syncthreads();
    for (int off = 128; off > 0; off >>= 1) {
        if (threadIdx.x < off) s[threadIdx.x] += s[threadIdx.x + off];
        __syncthreads();
    }
    if (threadIdx.x == 0) o[row] = s[0];
}

// ---------------- tens = u_i + v_j - 2*M ----------------
__global__ void k_tens(const float* __restrict__ u, const float* __restrict__ v,
                       const float* __restrict__ Mt, float* __restrict__ D, int n) {
    int j = blockIdx.x * blockDim.x + threadIdx.x;
    int i = blockIdx.y;
    size_t idx = (size_t)i * n + j;
    D[idx] = u[i] + v[j] - 2.0f * Mt[idx];
}

// ---------------- soft-DTW forward wavefront (single 1024-thread WG) ----------------
// Thread i owns row i. Cell (i, j=d-i) address collapses to base + d with
// base = i*(NDIM-1), so the inner loop is immediate-offset loads around one VGPR.
__global__ __launch_bounds__(1024) void k_sdtw_fwd(const float* __restrict__ D,
                                                   float* __restrict__ R) {
    __shared__ float Rbuf[3][NDIM];
    const int i = threadIdx.x;
    const float* Db = D + (size_t)i * (NDIM - 1);
    float* Rb = R + (size_t)i * (NDIM - 1);
    Rbuf[0][i] = BIGV; Rbuf[1][i] = BIGV; Rbuf[2][i] = BIGV;
    __syncthreads();
    const int dlo = i, dhi = i + NDIM;           // valid d range: j in [0, NDIM)
    for (int d = 0; d < 2 * NDIM - 1; ++d) {
        const float* Rp  = Rbuf[(d + 2) % 3];    // diag d-1
        const float* Rpp = Rbuf[(d + 1) % 3];    // diag d-2
        float* Rc = Rbuf[d % 3];
        float rd = BIGV;
        if (d >= dlo && d < dhi) {
            float r_left = Rp[i];
            float r_up   = (i > 0) ? Rp[i - 1] : BIGV;
            float r_diag = (i > 0) ? Rpp[i - 1] : ((d == 0) ? 0.0f : BIGV);
            float zmin = fminf(r_left, fminf(r_up, r_diag));
            float s = __expf((zmin - r_left) * INVG) + __expf((zmin - r_up) * INVG) +
                      __expf((zmin - r_diag) * INVG);
            rd = Db[d] + (zmin - GAMMA * __logf(s));
            Rb[d] = rd;
        }
        Rc[i] = rd;
        __syncthreads();
    }
}

// ---------------- soft-DTW backward: A <- E = d softdtw / d D ----------------
// Neighbors (i+1,j), (i,j+1), (i+1,j+1) are base + d + {NDIM, 1, NDIM+1}.
__global__ __launch_bounds__(1024) void k_sdtw_bwd(const float* __restrict__ D,
                                                   const float* __restrict__ R,
                                                   float* __restrict__ A) {
    __shared__ float Ebuf[3][NDIM];
    const int i = threadIdx.x;
    const size_t base = (size_t)i * (NDIM - 1);
    Ebuf[0][i] = 0.0f; Ebuf[1][i] = 0.0f; Ebuf[2][i] = 0.0f;
    __syncthreads();
    const int dlo = i, dhi = i + NDIM;
    for (int d = 2 * NDIM - 2; d >= 0; --d) {
        const float* Ep  = Ebuf[(d + 1) % 3];    // diag d+1
        const float* Epp = Ebuf[(d + 2) % 3];    // diag d+2
        float* Ec = Ebuf[d % 3];
        float e = 0.0f;
        if (d >= dlo && d < dhi) {
            size_t a = base + d;
            if (i == NDIM - 1 && d == 2 * NDIM - 2) {
                e = 1.0f;
            } else {
                float rij = R[a];
                bool dn = (i + 1 < NDIM);            // (i+1, j) exists
                bool rt = (d + 1 < dhi);             // (i, j+1) exists
                if (dn)
                    e += Ep[i + 1] * __expf((R[a + NDIM] - rij - D[a + NDIM]) * INVG);
                if (rt)
                    e += Ep[i] * __expf((R[a + 1] - rij - D[a + 1]) * INVG);
                if (dn && rt)
                    e += Epp[i + 1] * __expf((R[a + NDIM + 1] - rij - D[a + NDIM + 1]) * INVG);
            }
            A[a] = e;
        }
        Ec[i] = e;
        __syncthreads();
    }
}

// ---------------- loss = sum(tens * A), deterministic two-stage ----------------
__global__ __launch_bounds__(256) void k_loss_rows(const float* __restrict__ D,
                                                   const float* __restrict__ A,
                                                   float* __restrict__ part, int n) {
    __shared__ float s[256];
    int row = blockIdx.x;
    float acc = 0.0f;
    for (int j = threadIdx.x; j < n; j += 256) {
        size_t idx = (size_t)row * n + j;
        acc += D[idx] * A[idx];
    }
    s[threadIdx.x] = acc; __syncthreads();
    for (int off = 128; off > 0; off >>= 1) {
        if (threadIdx.x < off) s[threadIdx.x] += s[threadIdx.x + off];
        __syncthreads();
    }
    if (threadIdx.x == 0) part[row] = s[0];
}

__global__ __launch_bounds__(1024) void k_loss_final(const float* __restrict__ part,
                                                     float* __restrict__ out, int n) {
    __shared__ float s[1024];
    int t = threadIdx.x;
    s[t] = (t < n) ? part[t] : 0.0f;
    __syncthreads();
    for (int off = 512; off > 0; off >>= 1) {
        if (t < off) s[t] += s[t + off];
        __syncthreads();
    }
    if (t == 0) out[0] = s[0];
}

extern "C" void kernel_launch(void* const* d_in, const int* in_sizes, int n_in,
                              void* d_out, int out_size, void* d_ws, size_t ws_size,
                              hipStream_t stream) {
    (void)in_sizes; (void)n_in; (void)out_size; (void)ws_size;
    const int n = NDIM, kdim = 64;
    const size_t NN = (size_t)n * n;
    const float* x = (const float*)d_in[0];
    const float* y = (const float*)d_in[1];

    float* ws = (float*)d_ws;
    float* C1 = ws;            // also holds Gram G1 before cost_finish
    float* C2 = C1 + NN;
    float* F1 = C2 + NN;
    float* F2 = F1 + NN;
    float* Amat = F2 + NN;
    float* Tmat = Amat + NN;
    float* Mmat = Tmat + NN;
    float* Dt   = Mmat + NN;   // tens
    float* Rmat = Dt + NN;     // soft-DTW R
    float* Ppart = Rmat + NN;  // 2*NN K-split partials
    float* xt = Ppart + 2 * NN;       // 64x1024
    float* yt = xt + (size_t)n * kdim;
    float* nx = yt + (size_t)n * kdim;
    float* ny = nx + n;
    float* p  = ny + n;
    float* q  = p + n;
    float* u  = q + n;
    float* v  = u + n;
    float* part = v + n;              // n floats (loss partials)
    float* cspart = part + n;         // 8*n floats (colsum partials)

    dim3 b256(256);
    dim3 gElem(n / 256, n);
    const int tilesTotal = (n / 32) * (n / 64);      // 512
    const int gemmBlocks = tilesTotal / 8;           // 64

    // cost matrices via Gram (K=64, no split)
    k_transpose<<<(n * kdim) / 256, b256, 0, stream>>>(x, xt, n, kdim);
    k_transpose<<<(n * kdim) / 256, b256, 0, stream>>>(y, yt, n, kdim);
    k_gemm_wmma<<<dim3(gemmBlocks, 1), b256, 0, stream>>>(x, xt, C1, n, n, kdim);
    k_gemm_wmma<<<dim3(gemmBlocks, 1), b256, 0, stream>>>(y, yt, C2, n, n, kdim);
    k_diag<<<n / 256, b256, 0, stream>>>(C1, nx, n);
    k_diag<<<n / 256, b256, 0, stream>>>(C2, ny, n);
    k_cost_finish<<<gElem, b256, 0, stream>>>(C1, F1, nx, n);
    k_cost_finish<<<gElem, b256, 0, stream>>>(C2, F2, ny, n);
    k_identity<<<gElem, b256, 0, stream>>>(Amat, n);

    for (int it = 0; it < 3; ++it) {
        k_rowsum<<<n, b256, 0, stream>>>(Amat, p, n);
        k_colsum_part<<<dim3(n / 256, 8), b256, 0, stream>>>(Amat, cspart, n);
        k_colsum_final<<<n / 256, b256, 0, stream>>>(cspart, q, n);
        k_matvec<<<n, b256, 0, stream>>>(F1, p, u, n);
        k_matvec<<<n, b256, 0, stream>>>(F2, q, v, n);
        // T = A @ C2  (K split in 2 inside one launch, then deterministic add)
        k_gemm_wmma<<<dim3(gemmBlocks, 2), b256, 0, stream>>>(Amat, C2, Ppart, n, n, n);
        k_add2<<<(int)(NN / 256), b256, 0, stream>>>(Ppart, Tmat, NN);
        // M = C1 @ T
        k_gemm_wmma<<<dim3(gemmBlocks, 2), b256, 0, stream>>>(C1, Tmat, Ppart, n, n, n);
        k_add2<<<(int)(NN / 256), b256, 0, stream>>>(Ppart, Mmat, NN);
        k_tens<<<gElem, b256, 0, stream>>>(u, v, Mmat, Dt, n);
        if (it == 2) {
            k_loss_rows<<<n, b256, 0, stream>>>(Dt, Amat, part, n);
            k_loss_final<<<1, 1024, 0, stream>>>(part, (float*)d_out, n);
        } else {
            k_sdtw_fwd<<<1, 1024, 0, stream>>>(Dt, Rmat);
            k_sdtw_bwd<<<1, 1024, 0, stream>>>(Dt, Rmat, Amat);
        }
    }
}